// DAGMAPostProcessingBlock_91096256348432
// MI455X (gfx1250) — compile-verified
//
#include <hip/hip_runtime.h>
#include <math.h>

// Problem constants (fixed by the reference)
static constexpr int N       = 128;
static constexpr int LD      = 132;   // padded LDS row stride (floats) -> conflict-free column reads
static constexpr int NB      = 8;     // 8x8 grid of 16x16 blocks
static constexpr int BS      = 16;
static constexpr int THREADS = 512;   // 16 wave32s
static constexpr int NWAVES  = THREADS / 32;
static constexpr int NITERS  = 1000;

static constexpr float S_PARAM   = 1.5f;
static constexpr float STEP_PRI  = 0.01f;
static constexpr float STEP_DUAL = 0.01f;
static constexpr float REG_SP    = 0.002f;
static constexpr float THRESH    = 0.5f;
static constexpr float N_LOG_S   = 128.0f * 0.40546510810816438f; // N*log(1.5)

typedef float v2f __attribute__((ext_vector_type(2)));
typedef float v8f __attribute__((ext_vector_type(8)));

// ---------------------------------------------------------------------------
// One wave computes a 16x16 block product with FP32 WMMA:
//   result = (ACC ? Cin : 0) + SIGN * A(16x16) * B(16x16)
// A,B,Cin are LDS pointers to the top-left of 16x16 tiles with row stride LD.
// K=16 is split into 4 chained v_wmma_f32_16x16x4_f32 ops.
// Fragment layouts per CDNA5 ISA 7.12.2 (wave32):
//   A 16x4  : lanes 0-15 -> K={kk,kk+1}, lanes 16-31 -> K={kk+2,kk+3}, M = lane&15
//   B 4x16  : VGPR v holds row K=kk+2*half+v, N = lane&15
//   C 16x16 : VGPR r holds rows r (lanes 0-15) and r+8 (lanes 16-31), N = lane&15
// SIGN is folded into the A fragment in VALU (f32 WMMA NEG bits are C-negate only).
// ---------------------------------------------------------------------------
template <int SIGN, bool ACC>
__device__ inline v8f wmma_block16(const float* __restrict__ A,
                                   const float* __restrict__ B,
                                   const float* __restrict__ Cin,
                                   int lane) {
  const int n = lane & 15;
  const int h = lane >> 4;
  v8f c;
  if (ACC) {
#pragma unroll
    for (int r = 0; r < 8; ++r) c[r] = Cin[(r + 8 * h) * LD + n];
  } else {
#pragma unroll
    for (int r = 0; r < 8; ++r) c[r] = 0.0f;
  }
#pragma unroll
  for (int kk = 0; kk < 16; kk += 4) {
    v2f a, b;
    a[0] = (float)SIGN * A[n * LD + kk + 2 * h + 0];
    a[1] = (float)SIGN * A[n * LD + kk + 2 * h + 1];
    b[0] = B[(kk + 2 * h + 0) * LD + n];
    b[1] = B[(kk + 2 * h + 1) * LD + n];
    // (neg_a, A, neg_b, B, c_mod, C, reuse_a, reuse_b)
    c = __builtin_amdgcn_wmma_f32_16x16x4_f32(false, a, false, b, (short)0, c,
                                              false, false);
  }
  return c;
}

__device__ inline void store_block16(float* __restrict__ C, v8f c, int lane) {
  const int n = lane & 15;
  const int h = lane >> 4;
#pragma unroll
  for (int r = 0; r < 8; ++r) C[(r + 8 * h) * LD + n] = c[r];
}

// ---------------------------------------------------------------------------
// One workgroup per batch. Entire 1000-iteration chain runs in-kernel; the
// 128x128 state x and working matrix S live in LDS.
// ---------------------------------------------------------------------------
__global__ __launch_bounds__(THREADS)
void dagma_postproc_kernel(const float* __restrict__ adj, float* __restrict__ out) {
  extern __shared__ float smem[];
  float* xs   = smem;               // N*LD : current x
  float* Ss   = smem + N * LD;      // N*LD : S, overwritten in place by inv(S)
  float* scal = smem + 2 * N * LD;  // [0]=alpha, [1]=logabsdet accumulator

  const int tid   = threadIdx.x;
  const int lane  = tid & 31;
  const int wave  = tid >> 5;
  const int batch = blockIdx.x;
  const float* __restrict__ adjb = adj + (size_t)batch * N * N;

  // x0 = adj
  for (int e = tid; e < N * N; e += THREADS) {
    const int i = e >> 7, j = e & (N - 1);
    xs[i * LD + j] = adjb[e];
  }
  if (tid == 0) scal[0] = 0.0f;
  __syncthreads();

  for (int iter = 0; iter < NITERS; ++iter) {
    // ---- S = s*I - x∘x --------------------------------------------------
    for (int e = tid; e < N * N; e += THREADS) {
      const int i = e >> 7, j = e & (N - 1);
      const float xv = xs[i * LD + j];
      Ss[i * LD + j] = ((i == j) ? S_PARAM : 0.0f) - xv * xv;
    }
    if (tid == 0) scal[1] = 0.0f;
    __syncthreads();

    // ---- blocked in-place Gauss-Jordan inversion + logabsdet ------------
    for (int k = 0; k < NB; ++k) {
      float* Pk = Ss + (k * BS) * LD + k * BS;

      // (1) unblocked GJ inversion of the 16x16 diagonal block (no pivoting:
      //     S is diagonally dominant by construction). Pivots -> logabsdet.
      const int i2 = (tid >> 4) & 15, j2 = tid & 15;
      for (int p = 0; p < BS; ++p) {
        float piv = 1.0f, rowv = 0.0f, colv = 0.0f, own = 0.0f;
        if (tid < 256) {
          piv  = Pk[p * LD + p];
          rowv = Pk[p * LD + j2];
          colv = Pk[i2 * LD + p];
          own  = Pk[i2 * LD + j2];
        }
        __syncthreads();
        if (tid == 0) scal[1] += logf(fabsf(piv));
        if (tid < 256) {
          const float ip = 1.0f / piv;
          float v;
          if (i2 == p)      v = (j2 == p) ? ip : rowv * ip;
          else if (j2 == p) v = -colv * ip;
          else              v = own - colv * rowv * ip;
          Pk[i2 * LD + j2] = v;
        }
        __syncthreads();
      }

      // (2) row-panel scaling: S[k][j] = P * S[k][j]  (j != k), one wave per j
      if (wave < NB - 1) {
        const int j  = wave + (wave >= k ? 1 : 0);
        float* Bkj   = Ss + (k * BS) * LD + j * BS;
        const v8f c  = wmma_block16<1, false>(Pk, Bkj, nullptr, lane);
        store_block16(Bkj, c, lane);
      }
      __syncthreads();

      // (3) trailing update S[i][j] -= S[i][k]*S[k][j]  (49 block-GEMMs),
      //     column scaling S[i][k] = -S[i][k]*P computed in regs, stored
      //     after a barrier so every trailing GEMM still sees the old S[i][k].
      for (int t = wave; t < (NB - 1) * (NB - 1); t += NWAVES) {
        const int ii = t / (NB - 1), jj = t % (NB - 1);
        const int i  = ii + (ii >= k ? 1 : 0);
        const int j  = jj + (jj >= k ? 1 : 0);
        float* Cb    = Ss + (i * BS) * LD + j * BS;
        const v8f c  = wmma_block16<-1, true>(Ss + (i * BS) * LD + k * BS,
                                              Ss + (k * BS) * LD + j * BS,
                                              Cb, lane);
        store_block16(Cb, c, lane);
      }
      v8f colc = {};
      if (wave < NB - 1) {
        const int i = wave + (wave >= k ? 1 : 0);
        colc = wmma_block16<-1, false>(Ss + (i * BS) * LD + k * BS, Pk,
                                       nullptr, lane);
      }
      __syncthreads();
      if (wave < NB - 1) {
        const int i = wave + (wave >= k ? 1 : 0);
        store_block16(Ss + (i * BS) * LD + k * BS, colc, lane);
      }
      __syncthreads();
    }
    // Ss now holds inv(S); scal[1] holds logabsdet(S).

    // ---- elementwise primal update + dual ascent ------------------------
    const float h     = -scal[1] + N_LOG_S;
    const float alpha = scal[0];
    for (int e = tid; e < N * N; e += THREADS) {
      const int i = e >> 7, j = e & (N - 1);
      float sc = adjb[e];                       // scores = threshold(adj)
      sc = (sc > THRESH) ? sc : 0.0f;
      const float xv  = xs[i * LD + j];
      const float gh  = 2.0f * Ss[i * LD + j] * xv;
      const float g   = -sc + alpha * gh;
      const float til = xv - STEP_PRI * g;
      const float nx  = fmaxf(fabsf(til) - REG_SP * STEP_PRI, 0.0f);
      xs[i * LD + j]  = fminf(nx, 1.0f);        // 1 - relu(1 - nx)
    }
    __syncthreads();
    if (tid == 0) scal[0] = alpha + STEP_DUAL * h;
    // next iteration's post-build barrier orders this write vs. readers
  }

  // ---- output: threshold(x) ---------------------------------------------
  for (int e = tid; e < N * N; e += THREADS) {
    const int i = e >> 7, j = e & (N - 1);
    const float xv = xs[i * LD + j];
    out[(size_t)batch * N * N + e] = (xv > THRESH) ? xv : 0.0f;
  }
}

extern "C" void kernel_launch(void* const* d_in, const int* in_sizes, int n_in,
                              void* d_out, int out_size, void* d_ws, size_t ws_size,
                              hipStream_t stream) {
  const float* adj = (const float*)d_in[0];
  float* out = (float*)d_out;
  const int batches = in_sizes[0] / (N * N);  // = 16
  const size_t shmem = (size_t)(2 * N * LD + 2) * sizeof(float);  // ~132 KB
  dagma_postproc_kernel<<<batches, THREADS, shmem, stream>>>(adj, out);
}